// camera_transf_21096879357972
// MI455X (gfx1250) — compile-verified
//
#include <hip/hip_runtime.h>
#include <hip/hip_bf16.h>
#include <math.h>

// ---------------------------------------------------------------------------
// out[b] = E @ x[b], E = exp_se3(w, v, theta)  (4x4, fp32)
// Memory-bound: 512 MiB traffic -> ~22us at 23.3 TB/s. Math done with the
// native fp32 matrix op V_WMMA_F32_16X16X4_F32 (D16x16 = A16x4 * B4x16 + C):
//   A = [E;E;E;E] (16x4), B = [x0|x1|x2|x3] (4x16)  => D rows 0-3 hold E@x_j.
// Two WMMAs per iteration (8 matrices); D_A lanes 0-15 + D_B lanes 16-31 are
// merged so all stores are full-wave 32-lane b32 stores.
// ---------------------------------------------------------------------------

typedef __attribute__((ext_vector_type(2))) float v2f;
typedef __attribute__((ext_vector_type(8))) float v8f;

// Kernel 1: one thread builds the 4x4 SE(3) exponential into d_ws (16 floats).
__global__ void build_se3_kernel(const float* __restrict__ w,
                                 const float* __restrict__ v,
                                 const float* __restrict__ th,
                                 float* __restrict__ E) {
    if (threadIdx.x == 0 && blockIdx.x == 0) {
        const float wx = w[0], wy = w[1], wz = w[2];
        const float vx = v[0], vy = v[1], vz = v[2];
        const float theta = th[0];

        float W[3][3] = {{0.f, -wz,  wy},
                         { wz, 0.f, -wx},
                         {-wy,  wx, 0.f}};
        float W2[3][3];
        #pragma unroll
        for (int i = 0; i < 3; ++i)
            #pragma unroll
            for (int j = 0; j < 3; ++j)
                W2[i][j] = W[i][0]*W[0][j] + W[i][1]*W[1][j] + W[i][2]*W[2][j];

        const float s = sinf(theta);
        const float c = cosf(theta);
        const float omc = 1.0f - c;        // (1 - cos)
        const float tms = theta - s;       // (theta - sin)

        float R[3][3], Vm[3][3], t[3];
        #pragma unroll
        for (int i = 0; i < 3; ++i) {
            #pragma unroll
            for (int j = 0; j < 3; ++j) {
                const float I = (i == j) ? 1.0f : 0.0f;
                R[i][j]  = I + s * W[i][j] + omc * W2[i][j];
                Vm[i][j] = theta * I + omc * W[i][j] + tms * W2[i][j];
            }
            t[i] = Vm[i][0]*vx + Vm[i][1]*vy + Vm[i][2]*vz;
        }
        #pragma unroll
        for (int i = 0; i < 3; ++i) {
            E[i*4 + 0] = R[i][0];
            E[i*4 + 1] = R[i][1];
            E[i*4 + 2] = R[i][2];
            E[i*4 + 3] = t[i];
        }
        E[12] = 0.0f; E[13] = 0.0f; E[14] = 0.0f; E[15] = 1.0f;
    }
}

// Kernel 2: streaming batched 4x4 left-multiply via V_WMMA_F32_16X16X4_F32.
__global__ __launch_bounds__(256)
void se3_apply_kernel(const float* __restrict__ x,
                      const float* __restrict__ E,
                      float* __restrict__ out,
                      unsigned int nmat) {
    const unsigned lane          = threadIdx.x & 31u;
    const unsigned wavesPerBlk   = blockDim.x >> 5;
    const unsigned globalWave    = blockIdx.x * wavesPerBlk + (threadIdx.x >> 5);
    const unsigned totalWaves    = gridDim.x * wavesPerBlk;
    const bool     lo            = lane < 16u;

    // --- A operand: [E;E;E;E] (16x4 f32) per ISA layout:
    //   v0: lanes0-15 (M=lane,K=0), lanes16-31 (M=lane-16,K=2); v1: K=1 / K=3
    const unsigned arow = lane & 3u;            // stacked-E row = M % 4
    const unsigned k0   = lo ? 0u : 2u;
    v2f a;
    a.x = E[arow * 4u + k0];
    a.y = E[arow * 4u + k0 + 1u];

    // --- B gather offsets (4x16 f32, layout assumed symmetric to A):
    //   v0: lanes0-15 (K=0,N=lane), lanes16-31 (K=2,N=lane-16); v1: K=1 / K=3
    // element (K,N) of [x0|x1|x2|x3] = base + (N>>2)*16 + K*4 + (N&3)
    const unsigned n      = lane & 15u;
    const unsigned khalf  = lo ? 0u : 8u;                 // K=0/1 vs K=2/3
    const unsigned ldOff0 = (n >> 2) * 16u + (n & 3u) + khalf;
    const unsigned ldOff1 = ldOff0 + 4u;
    // --- store offset: lane L covers matrix j=L>>2 (0..7), col L&3, +r*4
    const unsigned stOff  = (lane >> 2) * 16u + (lane & 3u);

    const unsigned ngroups = nmat >> 3;                   // groups of 8 matrices
    for (unsigned g = globalWave; g < ngroups; g += totalWaves) {
        const unsigned baseA = g * 128u;                  // 8 matrices * 16 floats
        const unsigned baseB = baseA + 64u;

        v2f bA, bB;
        bA.x = x[baseA + ldOff0];
        bA.y = x[baseA + ldOff1];
        bB.x = x[baseB + ldOff0];
        bB.y = x[baseB + ldOff1];

        v8f cz = {};
        v8f dA = __builtin_amdgcn_wmma_f32_16x16x4_f32(
            false, a, false, bA, (short)0, cz, false, false);
        v8f dB = __builtin_amdgcn_wmma_f32_16x16x4_f32(
            false, a, false, bB, (short)0, cz, false, false);

        // D VGPR r: lanes0-15 = (M=r), lanes16-31 = (M=8+r) == row r (A repeats).
        // Merge group A (lanes 0-15) with group B (lanes 16-31): full-wave stores.
        #pragma unroll
        for (int r = 0; r < 4; ++r) {
            const float srv = lo ? dA[r] : dB[r];
            out[baseA + stOff + (unsigned)r * 4u] = srv;
        }
    }

    // --- scalar tail for nmat % 8 (not taken for B = 4194304) ---
    const unsigned tailStart = ngroups << 3;
    if (tailStart < nmat) {
        const unsigned tid      = blockIdx.x * blockDim.x + threadIdx.x;
        const unsigned nthreads = gridDim.x * blockDim.x;
        for (unsigned m = tailStart + tid; m < nmat; m += nthreads) {
            float xi[16];
            #pragma unroll
            for (int k = 0; k < 16; ++k) xi[k] = x[m * 16u + k];
            #pragma unroll
            for (int r = 0; r < 4; ++r)
                #pragma unroll
                for (int cc = 0; cc < 4; ++cc)
                    out[m * 16u + r * 4 + cc] =
                        E[r*4+0]*xi[0*4+cc] + E[r*4+1]*xi[1*4+cc] +
                        E[r*4+2]*xi[2*4+cc] + E[r*4+3]*xi[3*4+cc];
        }
    }
}

extern "C" void kernel_launch(void* const* d_in, const int* in_sizes, int n_in,
                              void* d_out, int out_size, void* d_ws, size_t ws_size,
                              hipStream_t stream) {
    const float* w  = (const float*)d_in[0];
    const float* v  = (const float*)d_in[1];
    const float* th = (const float*)d_in[2];
    const float* x  = (const float*)d_in[3];
    float* out = (float*)d_out;
    float* E   = (float*)d_ws;                  // 16 floats of scratch

    const unsigned nmat = (unsigned)(in_sizes[3] / 16);   // 4,194,304

    build_se3_kernel<<<1, 32, 0, stream>>>(w, v, th, E);

    const int threads = 256;                    // 8 wave32 per block
    const int blocks  = 2048;                   // 16384 waves, grid-stride
    se3_apply_kernel<<<blocks, threads, 0, stream>>>(x, E, out, nmat);
}